// FastFeedForward_1924145349188
// MI455X (gfx1250) — compile-verified
//
#include <hip/hip_runtime.h>
#include <math.h>

typedef __attribute__((ext_vector_type(2))) float v2f;
typedef __attribute__((ext_vector_type(8))) float v8f;

#define FFF_DEPTH   11
#define N_NODES     21
#define MAX_IDX     20
#define DIM         768

#define WIN_ROWS    22    // 21 real node rows + 1 zero row for N-tile padding
#define WIN_STRIDE  772   // floats per w_in node row in LDS  (banks: 4n+k, conflict-free)
#define WOUT_STRIDE 28    // floats per transposed w_out dim row (banks: 4n+k, conflict-free)
#define SCR_STRIDE  36    // per-token scratch stride (even -> 8B-aligned b64, banks 4*(9n)+k)

#define WAVES_PER_BLOCK 8
#define BLOCK_THREADS   256
#define TOK_PER_WAVE    16

#define LDS_WIN_FLOATS   (WIN_ROWS * WIN_STRIDE)                     // 16984
#define LDS_WOUT_FLOATS  (DIM * WOUT_STRIDE)                         // 21504
#define LDS_SCR_FLOATS   (WAVES_PER_BLOCK * 2 * 16 * SCR_STRIDE)     // 9216
#define LDS_TOTAL_FLOATS (LDS_WIN_FLOATS + LDS_WOUT_FLOATS + LDS_SCR_FLOATS) // ~191KB

__global__ void fff_fused_kernel(const float* __restrict__ x,
                                 const float* __restrict__ w_in,
                                 const float* __restrict__ w_out,
                                 float* __restrict__ out) {
  extern __shared__ float lds[];
  float* s_win  = lds;                      // [22][772] w_in[node][k], row 21 = zeros
  float* s_wout = lds + LDS_WIN_FLOATS;     // [768][28] w_out^T[d][node] (cols 21..27 zero)
  float* s_scr  = s_wout + LDS_WOUT_FLOATS; // per-wave: [16][36] logits + [16][36] coeff

  const int tid  = threadIdx.x;
  const int wave = tid >> 5;
  const int lane = tid & 31;
  const int h    = lane >> 4;   // wave half (selects K pair / M half per ISA layout)
  const int n    = lane & 15;   // M row (A) / N column (B,C,D)

  // ---- Stage weights into LDS (reads coalesced over k) ----
  for (int node = 0; node < N_NODES; ++node) {
    const float* wiRow = w_in  + node * DIM;
    const float* woRow = w_out + node * DIM;
    for (int k = tid; k < DIM; k += BLOCK_THREADS) {
      s_win [node * WIN_STRIDE + k]  = wiRow[k];
      s_wout[k * WOUT_STRIDE + node] = woRow[k];   // transpose for contiguous K reads
    }
  }
  for (int k = tid; k < DIM; k += BLOCK_THREADS) {
    s_win[N_NODES * WIN_STRIDE + k] = 0.0f;        // zero padding row (node 21)
#pragma unroll
    for (int node = N_NODES; node < 28; ++node) s_wout[k * WOUT_STRIDE + node] = 0.0f;
  }

  const int tokBase = (blockIdx.x * WAVES_PER_BLOCK + wave) * TOK_PER_WAVE;
  float* s_log = s_scr + wave * (2 * 16 * SCR_STRIDE);
  float* s_cf  = s_log + 16 * SCR_STRIDE;

  // zero coeff rows (slots 21..35 double as K-padding zeros for phase 3)
  for (int i = lane; i < 16 * SCR_STRIDE; i += 32) s_cf[i] = 0.0f;

  __syncthreads();

  // ---- Phase 1: logits[16 tok][21 nodes] = X * Win^T via V_WMMA_F32_16X16X4_F32 ----
  // A (16x4 f32): lane(h,n) holds X[tok n][k0+2h], X[tok n][k0+2h+1]
  // B (4x16 f32): lane(h,n) holds Win[node][k0+2h], Win[node][k0+2h+1]
  const float* xrow = x + (size_t)(tokBase + n) * DIM + 2 * h;
  const float* wb0  = s_win + n * WIN_STRIDE + 2 * h;
  const int    nd1  = 16 + n;
  // invalid node columns (21..31) read the broadcast zero row -> no mask VALU needed
  const float* wb1  = s_win + (nd1 <= MAX_IDX ? nd1 : N_NODES) * WIN_STRIDE + 2 * h;

  v8f acc0 = {};
  v8f acc1 = {};
#pragma unroll 16
  for (int k0 = 0; k0 < DIM; k0 += 4) {
    v2f a  = *(const v2f*)(xrow + k0);
    v2f b0 = *(const v2f*)(wb0 + k0);
    v2f b1 = *(const v2f*)(wb1 + k0);
    acc0 = __builtin_amdgcn_wmma_f32_16x16x4_f32(false, a, false, b0, (short)0, acc0, false, false);
    acc1 = __builtin_amdgcn_wmma_f32_16x16x4_f32(false, a, false, b1, (short)0, acc1, false, false);
  }

  // C layout: vgpr r, lane(h,n) -> logits[tok r+8h][node n (+16)]
#pragma unroll
  for (int r = 0; r < 8; ++r) {
    s_log[(r + 8 * h) * SCR_STRIDE + n] = acc0[r];
  }
  if (n < N_NODES - 16) {       // single exec toggle for all 5 valid upper-node columns
#pragma unroll
    for (int r = 0; r < 8; ++r) {
      s_log[(r + 8 * h) * SCR_STRIDE + 16 + n] = acc1[r];
    }
  }
  __syncthreads();

  // ---- Phase 2: per-token tree walk + exact-erf GELU scatter (lanes 0..15) ----
  if (h == 0) {
    const float* Lp = s_log + n * SCR_STRIDE;
    float*       Cp = s_cf  + n * SCR_STRIDE;
    int idx = 0;
#pragma unroll
    for (int d = 1; d < FFF_DEPTH; ++d) {
      float lg  = Lp[idx];                           // logit at parent node
      int   nxt = 2 * idx + 1 + (lg > 0.0f ? 1 : 0); // child select
      nxt = (nxt > MAX_IDX) ? MAX_IDX : nxt;         // reference clip semantics
      float g = 0.5f * lg * (1.0f + erff(lg * 0.70710678118654752440f));
      Cp[nxt] += g;                                  // coefficient lands on child node
      idx = nxt;
    }
  }
  __syncthreads();

  // ---- Phase 3: out[16 tok][768] = coeff * Wout via WMMA (K padded to 24 w/ zeros) ----
  const float* ca      = s_cf + n * SCR_STRIDE + 2 * h;
  float*       outBase = out + (size_t)tokBase * DIM;
#pragma unroll 2
  for (int dt = 0; dt < DIM / 16; ++dt) {
    const float* wb = s_wout + (dt * 16 + n) * WOUT_STRIDE + 2 * h;
    v8f acc = {};
#pragma unroll
    for (int k0 = 0; k0 < 24; k0 += 4) {
      v2f a = *(const v2f*)(ca + k0);
      v2f b = *(const v2f*)(wb + k0);
      acc = __builtin_amdgcn_wmma_f32_16x16x4_f32(false, a, false, b, (short)0, acc, false, false);
    }
#pragma unroll
    for (int r = 0; r < 8; ++r) {
      outBase[(r + 8 * h) * DIM + dt * 16 + n] = acc[r];
    }
  }
}

extern "C" void kernel_launch(void* const* d_in, const int* in_sizes, int n_in,
                              void* d_out, int out_size, void* d_ws, size_t ws_size,
                              hipStream_t stream) {
  (void)n_in; (void)d_ws; (void)ws_size; (void)out_size;
  const float* x  = (const float*)d_in[0];
  const float* wi = (const float*)d_in[1];
  const float* wo = (const float*)d_in[2];
  float* out = (float*)d_out;

  const int nTok   = in_sizes[0] / DIM;                       // 16384
  const int blocks = nTok / (WAVES_PER_BLOCK * TOK_PER_WAVE); // 128
  const size_t shmem = (size_t)LDS_TOTAL_FLOATS * sizeof(float);

  (void)hipFuncSetAttribute((const void*)fff_fused_kernel,
                            hipFuncAttributeMaxDynamicSharedMemorySize, (int)shmem);
  fff_fused_kernel<<<blocks, BLOCK_THREADS, shmem, stream>>>(x, wi, wo, out);
}